// GAT_54082228191479
// MI455X (gfx1250) — compile-verified
//
#include <hip/hip_runtime.h>
#include <hip/hip_bf16.h>
#include <math.h>

#define NN 20000
#define EE 320000
#define HH 8
#define DD 32
#define F  256          // HH*DD
#define LDSPITCH 260    // padded row stride (floats) to avoid LDS bank conflicts

typedef float v2f __attribute__((ext_vector_type(2)));
typedef float v8f __attribute__((ext_vector_type(8)));

// ---------------------------------------------------------------------------
// GEMM: C[NN,F] = A[NN,F] x W[F,F], exact f32 via V_WMMA_F32_16X16X4_F32.
// Block = 256 threads = 8 waves. Block owns a 16-row slab of A (staged in LDS,
// loaded from global exactly once); each wave computes two 16x16 output tiles,
// chaining 64 f32 WMMAs over K=256.
//
// A 16x4 f32 layout (ISA 7.12.2): lanes 0-15 -> K={0,1}, lanes 16-31 -> K={2,3},
// M = lane&15. B 4x16 mirrors it: lanes 0-15 -> rows K={0,1}, lanes 16-31 ->
// rows K={2,3}, N = lane&15. C/D: VGPR v holds M = v + 8*(lane>>4), N = lane&15.
// ---------------------------------------------------------------------------
__global__ __launch_bounds__(256) void gat_gemm_wmma(
    const float* __restrict__ A, const float* __restrict__ W,
    float* __restrict__ C) {
  __shared__ float As[16 * LDSPITCH];
  const int tid  = threadIdx.x;
  const int wave = tid >> 5;
  const int lane = tid & 31;
  const int m0   = blockIdx.x << 4;

  // Cooperative load of the 16x256 A slab (float4 per thread x4)
  for (int i = tid; i < 16 * 64; i += 256) {
    const int r  = i >> 6;
    const int c4 = (i & 63) << 2;
    const float4 v = *reinterpret_cast<const float4*>(A + (size_t)(m0 + r) * F + c4);
    float* p = &As[r * LDSPITCH + c4];
    p[0] = v.x; p[1] = v.y; p[2] = v.z; p[3] = v.w;
  }
  __syncthreads();

  const int hi = lane >> 4;   // half-wave select (K pair / output M half)
  const int l  = lane & 15;   // M for A, N for B/C

  for (int nt = wave; nt < 16; nt += 8) {
    const int n0 = nt << 4;
    v8f c = {};
    for (int k0 = 0; k0 < F; k0 += 4) {
      const int ka = k0 + (hi << 1);
      v2f a, b;
      a.x = As[l * LDSPITCH + ka];
      a.y = As[l * LDSPITCH + ka + 1];
      b.x = W[(size_t)ka * F + n0 + l];
      b.y = W[(size_t)(ka + 1) * F + n0 + l];
      // (neg_a, A, neg_b, B, c_mod, C, reuse_a, reuse_b)
      c = __builtin_amdgcn_wmma_f32_16x16x4_f32(false, a, false, b, (short)0, c,
                                                false, false);
    }
#pragma unroll
    for (int v = 0; v < 8; ++v)
      C[(size_t)(m0 + (hi << 3) + v) * F + n0 + l] = c[v];
  }
}

// el[n,h] = sum_d h[n,h,d]*al[h,d]; er likewise. One thread per (n,head).
__global__ void gat_attn_coeff(const float* __restrict__ h,
                               const float* __restrict__ al,
                               const float* __restrict__ ar,
                               float* __restrict__ el, float* __restrict__ er) {
  const int i = blockIdx.x * blockDim.x + threadIdx.x;
  if (i >= NN * HH) return;
  const int n = i >> 3, hd = i & 7;
  const float* hp  = h + (size_t)n * F + hd * DD;
  const float* alp = al + hd * DD;
  const float* arp = ar + hd * DD;
  float sl = 0.f, sr = 0.f;
#pragma unroll 8
  for (int d = 0; d < DD; ++d) {
    const float v = hp[d];
    sl += v * alp[d];
    sr += v * arp[d];
  }
  el[i] = sl;
  er[i] = sr;
}

// Per-call init: m = -inf, s = 0, agg = 0.
__global__ void gat_init(float* __restrict__ m, float* __restrict__ s,
                         float* __restrict__ agg) {
  const int i = blockIdx.x * blockDim.x + threadIdx.x;
  if (i < NN * HH) { m[i] = -INFINITY; s[i] = 0.f; }
  if (i < NN * F) agg[i] = 0.f;
}

// Monotone float atomic-max via signed/unsigned integer ordering (-inf init).
__device__ __forceinline__ void atomicMaxF32(float* addr, float v) {
  if (v >= 0.f)
    atomicMax(reinterpret_cast<int*>(addr), __float_as_int(v));
  else
    atomicMin(reinterpret_cast<unsigned int*>(addr),
              (unsigned int)__float_as_int(v));
}

// e = leaky_relu(el[src]+er[dst]); segment max into m[dst]. Thread per (edge,head).
__global__ void gat_edge_max(const float* __restrict__ el,
                             const float* __restrict__ er,
                             const int* __restrict__ src,
                             const int* __restrict__ dst,
                             float* __restrict__ e, float* __restrict__ m) {
  const int i = blockIdx.x * blockDim.x + threadIdx.x;
  if (i >= EE * HH) return;
  const int ed = i >> 3, hd = i & 7;
  const int sn = src[ed], dn = dst[ed];
  float v = el[sn * HH + hd] + er[dn * HH + hd];
  v = (v > 0.f) ? v : 0.2f * v;
  e[i] = v;
  atomicMaxF32(&m[dn * HH + hd], v);
}

// ex = exp(e - m[dst]); segment sum into s[dst]; e buffer overwritten with ex.
__global__ void gat_edge_exp(float* __restrict__ e, const float* __restrict__ m,
                             const int* __restrict__ dst, float* __restrict__ s) {
  const int i = blockIdx.x * blockDim.x + threadIdx.x;
  if (i >= EE * HH) return;
  const int ed = i >> 3, hd = i & 7;
  const int dn = dst[ed];
  const float v = __expf(e[i] - m[dn * HH + hd]);
  e[i] = v;
  atomicAdd(&s[dn * HH + hd], v);
}

// alpha = ex / s[dst]; agg[dst,h,:] += alpha * hfeat[src,h,:].
__global__ void gat_edge_agg(const float* __restrict__ ex,
                             const float* __restrict__ s,
                             const int* __restrict__ src,
                             const int* __restrict__ dst,
                             const float* __restrict__ hfeat,
                             float* __restrict__ agg) {
  const int i = blockIdx.x * blockDim.x + threadIdx.x;
  if (i >= EE * HH) return;
  const int ed = i >> 3, hd = i & 7;
  const int sn = src[ed], dn = dst[ed];
  const float alpha = ex[i] / s[dn * HH + hd];
  const float* hp = hfeat + (size_t)sn * F + hd * DD;
  float* op = agg + (size_t)dn * F + hd * DD;
#pragma unroll 8
  for (int d = 0; d < DD; ++d) atomicAdd(&op[d], alpha * hp[d]);
}

// out = elu(agg + b)
__global__ void gat_bias_elu(const float* __restrict__ agg,
                             const float* __restrict__ b,
                             float* __restrict__ out) {
  const int i = blockIdx.x * blockDim.x + threadIdx.x;
  if (i >= NN * F) return;
  const float v = agg[i] + b[i & (F - 1)];
  out[i] = (v > 0.f) ? v : (__expf(v) - 1.f);
}

extern "C" void kernel_launch(void* const* d_in, const int* in_sizes, int n_in,
                              void* d_out, int out_size, void* d_ws, size_t ws_size,
                              hipStream_t stream) {
  const float* feats = (const float*)d_in[0];
  const float* W0    = (const float*)d_in[1];
  const float* al0   = (const float*)d_in[2];
  const float* ar0   = (const float*)d_in[3];
  const float* b0    = (const float*)d_in[4];
  const float* W1    = (const float*)d_in[5];
  const float* al1   = (const float*)d_in[6];
  const float* ar1   = (const float*)d_in[7];
  const float* b1    = (const float*)d_in[8];
  const int*   src   = (const int*)d_in[9];
  const int*   dst   = (const int*)d_in[10];

  float* ws = (float*)d_ws;
  // Workspace layout (floats): ~74.2 MB total
  float* hbuf = ws;                         // [NN,F]  post-GEMM features
  float* agg  = hbuf + (size_t)NN * F;      // [NN,F]  aggregation target
  float* x1   = agg + (size_t)NN * F;       // [NN,F]  layer-1 output
  float* el   = x1 + (size_t)NN * F;        // [NN,HH]
  float* er   = el + (size_t)NN * HH;       // [NN,HH]
  float* mmax = er + (size_t)NN * HH;       // [NN,HH]
  float* ssum = mmax + (size_t)NN * HH;     // [NN,HH]
  float* ebuf = ssum + (size_t)NN * HH;     // [EE,HH] e -> ex (in place)

  const int bl = 256;
  const int gNF = (NN * F + bl - 1) / bl;
  const int gNH = (NN * HH + bl - 1) / bl;
  const int gEH = (EE * HH + bl - 1) / bl;

  auto run_layer = [&](const float* x, const float* Wm, const float* al,
                       const float* ar, const float* bb, float* outx) {
    gat_gemm_wmma<<<NN / 16, 256, 0, stream>>>(x, Wm, hbuf);
    gat_attn_coeff<<<gNH, bl, 0, stream>>>(hbuf, al, ar, el, er);
    gat_init<<<gNF, bl, 0, stream>>>(mmax, ssum, agg);
    gat_edge_max<<<gEH, bl, 0, stream>>>(el, er, src, dst, ebuf, mmax);
    gat_edge_exp<<<gEH, bl, 0, stream>>>(ebuf, mmax, dst, ssum);
    gat_edge_agg<<<gEH, bl, 0, stream>>>(ebuf, ssum, src, dst, hbuf, agg);
    gat_bias_elu<<<gNF, bl, 0, stream>>>(agg, bb, outx);
  };

  run_layer(feats, W0, al0, ar0, b0, x1);
  run_layer(x1, W1, al1, ar1, b1, (float*)d_out);
}